// RankingLoss_23201413333492
// MI455X (gfx1250) — compile-verified
//
#include <hip/hip_runtime.h>

typedef __attribute__((ext_vector_type(2))) float v2f;
typedef __attribute__((ext_vector_type(8))) float v8f;

#define MARGIN 1.0f
#define JTILE  1024
#define ISTRIP 128   // rows per block (8 waves * 16 rows)

__global__ void __launch_bounds__(64) rank_init(float* ws) {
    if (threadIdx.x == 0) {
        ws[0] = 0.0f;                       // loss accumulator (f32)
        ((unsigned int*)ws)[1] = 0u;        // pair-count accumulator (u32, exact)
    }
}

__global__ void __launch_bounds__(256) rank_main(const float* __restrict__ preds,
                                                 const float* __restrict__ targets,
                                                 float* __restrict__ ws) {
    __shared__ __align__(16) float2 s_pj[JTILE];   // {pred_j, dur_j}
    __shared__ float        s_loss[8];
    __shared__ unsigned int s_cnt[8];

    const int tid   = threadIdx.x;
    const int jbase = blockIdx.y * JTILE;

    // Stage this block's j-window into LDS as interleaved {pred, dur}.
    for (int t = tid; t < JTILE; t += 256) {
        const int j = jbase + t;
        s_pj[t] = make_float2(preds[j], targets[2 * j]);
    }
    __syncthreads();

    const int wave  = tid >> 5;          // 0..7
    const int lane  = tid & 31;
    const int m     = lane & 15;         // A-matrix row within 16-row strip
    const int khalf = lane >> 4;         // 0 -> K{0,1}, 1 -> K{2,3} (f32 A layout)
    const int i     = blockIdx.x * ISTRIP + wave * 16 + m;

    const float p_i = preds[i];
    const float d_i = targets[2 * i];
    const bool  e_i = (targets[2 * i + 1] == 1.0f);
    // Fold the event gate into the duration: !e_i => comparison always false.
    const float d_gate = e_i ? d_i : 3.0e38f;
    // Hoist loop-invariant part of the hinge: relu(MARGIN - (p_i - p_j))
    //   == max(c_i + p_j, 0) with c_i = MARGIN - p_i  (1 VALU/pair saved).
    const float c_i = MARGIN - p_i;

    const float4* __restrict__ s_pj4 = (const float4*)s_pj;

    v2f ones; ones.x = 1.0f; ones.y = 1.0f;  // B = all ones -> WMMA computes row sums
    v8f acc0 = {}, acc1 = {}, acc2 = {}, acc3 = {};
    unsigned int cnt = 0;                    // wave-uniform (ballot/popcount on SALU)

    // Per lane: build A[m][k], A[m][k+1] = masked hinge for pairs (i, j), (i, j+1).
    auto make_a = [&](int j) -> v2f {
        const float4 pj = s_pj4[j >> 1];     // pred_j, dur_j, pred_j1, dur_j1 (b128)
        const bool v0 = d_gate < pj.y;
        const bool v1 = d_gate < pj.w;
        const float h0 = fmaxf(c_i + pj.x, 0.0f);
        const float h1 = fmaxf(c_i + pj.z, 0.0f);
        v2f a;
        a.x = v0 ? h0 : 0.0f;
        a.y = v1 ? h1 : 0.0f;
        cnt += (unsigned)__builtin_popcount(__builtin_amdgcn_ballot_w32(v0));
        cnt += (unsigned)__builtin_popcount(__builtin_amdgcn_ballot_w32(v1));
        return a;
    };

    for (int j0 = 0; j0 < JTILE; j0 += 16) {
        const int jb = j0 + khalf * 2;
        {   v2f a = make_a(jb + 0);
            acc0 = __builtin_amdgcn_wmma_f32_16x16x4_f32(false, a, false, ones,
                                                         (short)0, acc0, false, false); }
        {   v2f a = make_a(jb + 4);
            acc1 = __builtin_amdgcn_wmma_f32_16x16x4_f32(false, a, false, ones,
                                                         (short)0, acc1, false, false); }
        {   v2f a = make_a(jb + 8);
            acc2 = __builtin_amdgcn_wmma_f32_16x16x4_f32(false, a, false, ones,
                                                         (short)0, acc2, false, false); }
        {   v2f a = make_a(jb + 12);
            acc3 = __builtin_amdgcn_wmma_f32_16x16x4_f32(false, a, false, ones,
                                                         (short)0, acc3, false, false); }
    }

    // Columns of D are identical; lanes 0-15 hold rows 0-7 across the 8 VGPRs,
    // lanes 16-31 hold rows 8-15. Wave total = lane0 + lane16 of per-lane VGPR sum.
    const v8f acc = acc0 + acc1 + acc2 + acc3;
    float s = acc[0] + acc[1] + acc[2] + acc[3] + acc[4] + acc[5] + acc[6] + acc[7];
    const float wloss = __shfl(s, 0, 32) + __shfl(s, 16, 32);

    if (lane == 0) { s_loss[wave] = wloss; s_cnt[wave] = cnt; }
    __syncthreads();

    if (tid == 0) {
        float L = 0.0f; unsigned int C = 0u;
        for (int w = 0; w < 8; ++w) { L += s_loss[w]; C += s_cnt[w]; }
        atomicAdd(&ws[0], L);
        atomicAdd(((unsigned int*)ws) + 1, C);
    }
}

__global__ void __launch_bounds__(64) rank_fin(const float* __restrict__ ws,
                                               float* __restrict__ out) {
    if (threadIdx.x == 0) {
        const unsigned int c = ((const unsigned int*)ws)[1];
        out[0] = (c > 0u) ? (ws[0] / (float)c) : 0.0f;
    }
}

extern "C" void kernel_launch(void* const* d_in, const int* in_sizes, int n_in,
                              void* d_out, int out_size, void* d_ws, size_t ws_size,
                              hipStream_t stream) {
    (void)n_in; (void)out_size; (void)ws_size;
    const float* preds   = (const float*)d_in[0];
    const float* targets = (const float*)d_in[1];   // (N,2) flattened: dur, event
    float* out = (float*)d_out;
    float* ws  = (float*)d_ws;

    const int n  = in_sizes[0];        // 8192
    const int gx = n / ISTRIP;         // 64 i-strips
    const int gy = n / JTILE;          // 8 j-splits

    rank_init<<<1, 64, 0, stream>>>(ws);
    rank_main<<<dim3(gx, gy), 256, 0, stream>>>(preds, targets, ws);
    rank_fin<<<1, 64, 0, stream>>>(ws, out);
}